// GATModel_18098992185491
// MI455X (gfx1250) — compile-verified
//
#include <hip/hip_runtime.h>
#include <cstddef>

namespace {

constexpr int BS = 256;
constexpr float SLOPE = 0.2f;

typedef float v2f __attribute__((ext_vector_type(2)));
typedef float v8f __attribute__((ext_vector_type(8)));

__device__ __forceinline__ float lrelu(float x) { return x > 0.0f ? x : SLOPE * x; }

// float atomic max via signed/unsigned int ordering (valid incl. negatives,
// init value -FLT_MAX).
__device__ __forceinline__ void atomicMaxF(float* addr, float v) {
  if (v >= 0.0f) atomicMax((int*)addr, __float_as_int(v));
  else           atomicMin((unsigned int*)addr, __float_as_uint(v));
}

__global__ void fill_kernel(float* __restrict__ p, float v, int n) {
  int i = blockIdx.x * blockDim.x + threadIdx.x;
  if (i < n) p[i] = v;
}

// One wave computes one 16x16 tile of Y = X[MxK] * W[KxNCOL] using f32 WMMA.
// K, NCOL compile-time so the unrolled loop uses immediate-offset global loads.
// A layout (16x4 f32): lane m=lane&15; half-wave selects K pair; vgpr = k&1.
// B layout (4x16 f32): lane n=lane&15; half-wave selects K pair; vgpr = k&1.
// C/D layout (16x16 f32): vgpr r -> row r (lanes 0-15) / row r+8 (lanes 16-31).
template <int K, int NCOL>
__global__ void gemm_f32_wmma(const float* __restrict__ X, const float* __restrict__ W,
                              float* __restrict__ Y, int M) {
  constexpr int NT = NCOL / 16;
  const int wavesPerBlock = blockDim.x >> 5;
  const int waveId = blockIdx.x * wavesPerBlock + ((int)threadIdx.x >> 5);
  const int mtile = waveId / NT;
  const int ntile = waveId - mtile * NT;
  if (mtile * 16 >= M) return;  // wave-uniform: EXEC stays all-ones for WMMA
  const int lane = (int)threadIdx.x & 31;
  const int lhalf = lane >> 4;   // selects K pair {0,1} vs {2,3}
  const int lmod = lane & 15;

  // 8-byte aligned: (row*K + 2*lhalf)*4 is a multiple of 8.
  const float* xrow = X + (size_t)(mtile * 16 + lmod) * K + 2 * lhalf;
  const float* wcol = W + (size_t)(2 * lhalf) * NCOL + ntile * 16 + lmod;

  v8f acc = {};
#pragma unroll
  for (int k0 = 0; k0 < K; k0 += 4) {
    v2f a = *(const v2f*)(xrow + k0);
    v2f b;
    b.x = wcol[k0 * NCOL];
    b.y = wcol[k0 * NCOL + NCOL];
    acc = __builtin_amdgcn_wmma_f32_16x16x4_f32(false, a, false, b, (short)0, acc,
                                                false, false);
  }

  float* out = Y + (size_t)(mtile * 16 + lhalf * 8) * NCOL + ntile * 16 + lmod;
#pragma unroll
  for (int r = 0; r < 8; ++r) out[r * NCOL] = acc[r];
}

// e_src[n,h] = sum_c h[n,h,c]*a_s[h,c];  e_dst likewise.
template <int H, int C>
__global__ void attn_scores(const float* __restrict__ h, const float* __restrict__ a_s,
                            const float* __restrict__ a_d, float* __restrict__ es,
                            float* __restrict__ ed, int N) {
  int i = blockIdx.x * blockDim.x + threadIdx.x;
  if (i >= N * H) return;
  int n = i / H;
  int hh = i - n * H;
  const float* hv = h + (size_t)n * (H * C) + hh * C;
  float s = 0.f, d = 0.f;
#pragma unroll
  for (int c = 0; c < C; ++c) {
    float v = hv[c];
    s += v * a_s[hh * C + c];
    d += v * a_d[hh * C + c];
  }
  es[i] = s;
  ed[i] = d;
}

template <int H>
__global__ void edge_max(const long long* __restrict__ srcI, const long long* __restrict__ dstI,
                         int E, int Etot, const float* __restrict__ es,
                         const float* __restrict__ ed, float* __restrict__ mx) {
  int e = blockIdx.x * blockDim.x + threadIdx.x;
  if (e >= Etot) return;
  int s = (e < E) ? (int)srcI[e] : (e - E);
  int d = (e < E) ? (int)dstI[e] : (e - E);
#pragma unroll
  for (int h = 0; h < H; ++h) {
    float v = lrelu(es[s * H + h] + ed[d * H + h]);
    atomicMaxF(&mx[d * H + h], v);
  }
}

template <int H>
__global__ void edge_denom(const long long* __restrict__ srcI, const long long* __restrict__ dstI,
                           int E, int Etot, const float* __restrict__ es,
                           const float* __restrict__ ed, const float* __restrict__ mx,
                           float* __restrict__ dn) {
  int e = blockIdx.x * blockDim.x + threadIdx.x;
  if (e >= Etot) return;
  int s = (e < E) ? (int)srcI[e] : (e - E);
  int d = (e < E) ? (int)dstI[e] : (e - E);
#pragma unroll
  for (int h = 0; h < H; ++h) {
    float v = lrelu(es[s * H + h] + ed[d * H + h]);
    atomicAdd(&dn[d * H + h], __expf(v - mx[d * H + h]));
  }
}

// One thread per (edge, channel): accum[dst, c] += alpha[e,h] * hlin[src, c]
template <int H, int C>
__global__ void edge_aggr(const long long* __restrict__ srcI, const long long* __restrict__ dstI,
                          int E, int Etot, const float* __restrict__ es,
                          const float* __restrict__ ed, const float* __restrict__ mx,
                          const float* __restrict__ dn, const float* __restrict__ hlin,
                          float* __restrict__ accum) {
  constexpr int HC = H * C;
  long long tid = (long long)blockIdx.x * blockDim.x + threadIdx.x;
  if (tid >= (long long)Etot * HC) return;
  int e = (int)(tid / HC);
  int c = (int)(tid - (long long)e * HC);
  int h = c / C;
  int s = (e < E) ? (int)srcI[e] : (e - E);
  int d = (e < E) ? (int)dstI[e] : (e - E);
  float v = lrelu(es[s * H + h] + ed[d * H + h]);
  float alpha = __expf(v - mx[d * H + h]) / (dn[d * H + h] + 1e-16f);
  atomicAdd(&accum[(size_t)d * HC + c], alpha * hlin[(size_t)s * HC + c]);
}

__global__ void bias_act_kernel(const float* __restrict__ acc, const float* __restrict__ bias,
                                float* __restrict__ out, int total, int HC, int doRelu) {
  int i = blockIdx.x * blockDim.x + threadIdx.x;
  if (i >= total) return;
  int c = i % HC;
  float v = acc[i] + bias[c];
  if (doRelu) v = fmaxf(v, 0.0f);
  out[i] = v;
}

static inline unsigned int cdiv(long long a, int b) { return (unsigned int)((a + b - 1) / b); }

template <int K, int H, int C>
static void launch_layer(const float* xin, const float* W, const float* a_s,
                         const float* a_d, const float* bias, float* hlin, float* accum,
                         float* out, bool doRelu, float* es, float* ed, float* mx, float* dn,
                         const long long* srcI, const long long* dstI, int N, int E, int Etot,
                         hipStream_t stream) {
  constexpr int HC = H * C;
  {  // h = x @ W  (f32 WMMA)
    long long waves = (long long)((N + 15) / 16) * (HC / 16);
    gemm_f32_wmma<K, HC><<<cdiv(waves * 32, BS), BS, 0, stream>>>(xin, W, hlin, N);
  }
  attn_scores<H, C><<<cdiv((long long)N * H, BS), BS, 0, stream>>>(hlin, a_s, a_d, es, ed, N);
  fill_kernel<<<cdiv((long long)N * H, BS), BS, 0, stream>>>(mx, -3.402823e38f, N * H);
  fill_kernel<<<cdiv((long long)N * H, BS), BS, 0, stream>>>(dn, 0.0f, N * H);
  fill_kernel<<<cdiv((long long)N * HC, BS), BS, 0, stream>>>(accum, 0.0f, N * HC);
  edge_max<H><<<cdiv(Etot, BS), BS, 0, stream>>>(srcI, dstI, E, Etot, es, ed, mx);
  edge_denom<H><<<cdiv(Etot, BS), BS, 0, stream>>>(srcI, dstI, E, Etot, es, ed, mx, dn);
  edge_aggr<H, C><<<cdiv((long long)Etot * HC, BS), BS, 0, stream>>>(srcI, dstI, E, Etot, es, ed,
                                                                     mx, dn, hlin, accum);
  bias_act_kernel<<<cdiv((long long)N * HC, BS), BS, 0, stream>>>(
      accum, bias, out, N * HC, HC, doRelu ? 1 : 0);
}

}  // namespace

extern "C" void kernel_launch(void* const* d_in, const int* in_sizes, int n_in,
                              void* d_out, int out_size, void* d_ws, size_t ws_size,
                              hipStream_t stream) {
  const float* x  = (const float*)d_in[0];
  const long long* ei = (const long long*)d_in[1];
  const float* W1  = (const float*)d_in[2];
  const float* as1 = (const float*)d_in[3];
  const float* ad1 = (const float*)d_in[4];
  const float* b1  = (const float*)d_in[5];
  const float* W2  = (const float*)d_in[6];
  const float* as2 = (const float*)d_in[7];
  const float* ad2 = (const float*)d_in[8];
  const float* b2  = (const float*)d_in[9];
  const float* W3  = (const float*)d_in[10];
  const float* as3 = (const float*)d_in[11];
  const float* ad3 = (const float*)d_in[12];
  const float* b3  = (const float*)d_in[13];

  const int IN = 128, HID = 32, HEADS = 4;
  const int HC = HEADS * HID;  // 128
  const int N = in_sizes[0] / IN;
  const int E = in_sizes[1] / 2;
  const int Etot = E + N;
  const long long* srcI = ei;       // edge_index[0]
  const long long* dstI = ei + E;   // edge_index[1]

  float* ws = (float*)d_ws;
  float* A  = ws;                       // hlin buffer        N*HC
  float* B  = A + (size_t)N * HC;       // layer-1 activation N*HC
  float* C_ = B + (size_t)N * HC;       // layer-2 activation N*HC
  float* es = C_ + (size_t)N * HC;      // N*HEADS
  float* ed = es + (size_t)N * HEADS;   // N*HEADS
  float* mx = ed + (size_t)N * HEADS;   // N*HEADS
  float* dn = mx + (size_t)N * HEADS;   // N*HEADS
  float* out = (float*)d_out;

  // Layer 1: x[N,128] -> B[N,128], relu
  launch_layer<128, 4, 32>(x, W1, as1, ad1, b1, A, B, B, /*relu=*/true,
                           es, ed, mx, dn, srcI, dstI, N, E, Etot, stream);
  // Layer 2: B[N,128] -> C_[N,128], relu
  launch_layer<128, 4, 32>(B, W2, as2, ad2, b2, A, C_, C_, /*relu=*/true,
                           es, ed, mx, dn, srcI, dstI, N, E, Etot, stream);
  // Layer 3: C_[N,128] -> d_out[N,16], no relu (reuses A for hlin, accum=d_out)
  launch_layer<128, 1, 16>(C_, W3, as3, ad3, b3, A, out, out, /*relu=*/false,
                           es, ed, mx, dn, srcI, dstI, N, E, Etot, stream);
}